// MyMultihead_33440615366831
// MI455X (gfx1250) — compile-verified
//
#include <hip/hip_runtime.h>

// MI455X / gfx1250: wave32, f32 WMMA (16x16x4) path.
// Register-blocked 64x64-per-wave GEMMs; attention kernels hoist
// loop-invariant fragments and use a transposed V layout for b64 B-frag loads.

typedef __attribute__((ext_vector_type(2))) float v2f;
typedef __attribute__((ext_vector_type(4))) float v4f;
typedef __attribute__((ext_vector_type(8))) float v8f;

#define BB    2
#define SS    2048
#define EE    1024
#define NHEAD 16
#define HH    64

static __device__ __forceinline__ v8f wmma_f32(v2f a, v2f b, v8f c) {
  // 8 args: (neg_a, A, neg_b, B, c_mod, C, reuse_a, reuse_b)
  return __builtin_amdgcn_wmma_f32_16x16x4_f32(false, a, false, b, (short)0, c,
                                               false, false);
}

// ---------------------------------------------------------------------------
// Kernel 1: per-head projection  P[b,n,s,h] = sum_e X[b,s,e]*W[n,e,h] + bias
// One wave = 64 rows (s) x 64 cols (h == full H) = 4x4 WMMA tiles.
// Per K-step: 4 b64 A-loads + 8 b32 B-loads feed 16 WMMAs.
// store_transposed != 0 writes P as [b,n,h,s] (for the V projection) with
// per-lane contiguous float4 stores.
// ---------------------------------------------------------------------------
__global__ __launch_bounds__(256) void mha_proj_kernel(
    const float* __restrict__ X,     // [B,S,E]
    const float* __restrict__ W,     // [N,E,H]
    const float* __restrict__ bias,  // [N,1,H]
    float* __restrict__ P,           // [B,N,S,H] or [B,N,H,S]
    int store_transposed)
{
  const int wave = (blockIdx.x * blockDim.x + threadIdx.x) >> 5;
  const int lane = threadIdx.x & 31;
  const int SB = SS / 64;              // 32 row-blocks
  const int sb = wave % SB;
  const int n  = (wave / SB) % NHEAD;
  const int b  = wave / (SB * NHEAD);

  const int m     = lane & 15;         // A-row / B-col within tile
  const int kp    = (lane >> 4) << 1;  // K sub-offset {0,2}
  const int rbase = (lane >> 4) << 3;  // D-row base {0,8}

  const float* Xrow0 = X + ((size_t)b * SS + sb * 64 + m) * EE;
  const float* Wbase = W + (size_t)n * EE * HH + m;

  v8f acc[4][4] = {};
  for (int e = 0; e < EE; e += 4) {
    v2f a[4];
#pragma unroll
    for (int rt = 0; rt < 4; ++rt)
      a[rt] = *(const v2f*)(Xrow0 + (size_t)rt * 16 * EE + e + kp);
#pragma unroll
    for (int ct = 0; ct < 4; ++ct) {
      v2f bb;
      bb.x = Wbase[(size_t)(e + kp) * HH + ct * 16];
      bb.y = Wbase[(size_t)(e + kp + 1) * HH + ct * 16];
#pragma unroll
      for (int rt = 0; rt < 4; ++rt)
        acc[rt][ct] = wmma_f32(a[rt], bb, acc[rt][ct]);
    }
  }

  const float* biasn = bias + n * HH + m;
  if (!store_transposed) {
    // P[b,n,s,h]
    float* Pbase = P + (((size_t)b * NHEAD + n) * SS + sb * 64) * HH + m;
#pragma unroll
    for (int rt = 0; rt < 4; ++rt)
#pragma unroll
      for (int ct = 0; ct < 4; ++ct) {
        const float bval = biasn[ct * 16];
#pragma unroll
        for (int r = 0; r < 8; ++r)
          Pbase[(size_t)(rt * 16 + rbase + r) * HH + ct * 16] =
              acc[rt][ct][r] + bval;
      }
  } else {
    // P[b,n,h,s]: per-lane contiguous, 16B-aligned float4 stores
    float* Tbase = P + (((size_t)b * NHEAD + n) * HH + m) * SS + sb * 64 + rbase;
#pragma unroll
    for (int ct = 0; ct < 4; ++ct) {
      const float bval = biasn[ct * 16];
#pragma unroll
      for (int rt = 0; rt < 4; ++rt) {
        v4f lo = {acc[rt][ct][0] + bval, acc[rt][ct][1] + bval,
                  acc[rt][ct][2] + bval, acc[rt][ct][3] + bval};
        v4f hi = {acc[rt][ct][4] + bval, acc[rt][ct][5] + bval,
                  acc[rt][ct][6] + bval, acc[rt][ct][7] + bval};
        *(v4f*)(Tbase + (size_t)(ct * 16) * SS + rt * 16)     = lo;
        *(v4f*)(Tbase + (size_t)(ct * 16) * SS + rt * 16 + 4) = hi;
      }
    }
  }
}

// ---------------------------------------------------------------------------
// Kernel 2: online per-KEY-column max & exp-sum over the QUERY axis.
// One wave = one (b, n, k-tile). K-fragments are hoisted (invariant over qt).
// ---------------------------------------------------------------------------
__global__ __launch_bounds__(256) void mha_softmax_stats_kernel(
    const float* __restrict__ QP, const float* __restrict__ KP,
    float* __restrict__ colmax, float* __restrict__ colsum)
{
  const int wave = (blockIdx.x * blockDim.x + threadIdx.x) >> 5;
  const int lane = threadIdx.x & 31;
  const int KT = SS / 16;
  const int kt = wave % KT;
  const int n  = (wave / KT) % NHEAD;
  const int b  = wave / (KT * NHEAD);

  const int m  = lane & 15;
  const int kp = (lane >> 4) << 1;

  const size_t bn = (size_t)b * NHEAD + n;
  const float* Krow = KP + (bn * SS + kt * 16 + m) * HH;

  v2f kf[16];
#pragma unroll
  for (int h4 = 0; h4 < 16; ++h4)
    kf[h4] = *(const v2f*)(Krow + h4 * 4 + kp);

  float runmax = -3.0e38f, runsum = 0.0f;
  for (int qt = 0; qt < SS / 16; ++qt) {
    const float* Qrow = QP + (bn * SS + qt * 16 + m) * HH;
    v8f s = {};
#pragma unroll
    for (int h4 = 0; h4 < 16; ++h4) {
      v2f a = *(const v2f*)(Qrow + h4 * 4 + kp);
      s = wmma_f32(a, kf[h4], s);
    }
    float vals[8];
    float tmax = -3.0e38f;
#pragma unroll
    for (int r = 0; r < 8; ++r) {
      vals[r] = s[r] * 0.125f;                        // / sqrt(H)
      tmax = fmaxf(tmax, vals[r]);
    }
    tmax = fmaxf(tmax, __shfl_xor(tmax, 16, 32));     // combine row-halves
    const float newmax = fmaxf(runmax, tmax);
    float tsum = 0.0f;
#pragma unroll
    for (int r = 0; r < 8; ++r) tsum += __expf(vals[r] - newmax);
    tsum += __shfl_xor(tsum, 16, 32);
    runsum = runsum * __expf(runmax - newmax) + tsum;
    runmax = newmax;
  }
  if (lane < 16) {
    const size_t idx = bn * SS + kt * 16 + m;
    colmax[idx] = runmax;
    colsum[idx] = runsum;
  }
}

// ---------------------------------------------------------------------------
// Kernel 3: heads[b,q,n*H+v] = sum_k softmax_q(scores)[q,k] * vwv[b,n,k,v].
// One wave = (b, n, q-tile). Q-fragments hoisted (invariant over kt).
// Probability tile round-trips through LDS (D-layout -> A-layout).
// V is read from the transposed layout VPT[b,n,v,k] -> aligned b64 B-frags.
// ---------------------------------------------------------------------------
__global__ __launch_bounds__(256) void mha_heads_kernel(
    const float* __restrict__ QP, const float* __restrict__ KP,
    const float* __restrict__ VPT,   // [B,N,H,S] (v-major, k-contiguous)
    const float* __restrict__ colmax, const float* __restrict__ colsum,
    float* __restrict__ HS)          // [B,S,N*H]
{
  __shared__ float lds[8][16 * 18];  // per-wave 16x16 tile, row stride 18
  const int wave = (blockIdx.x * blockDim.x + threadIdx.x) >> 5;
  const int lane = threadIdx.x & 31;
  const int w    = (threadIdx.x >> 5) & 7;
  const int QT = SS / 16;
  const int qt = wave % QT;
  const int n  = (wave / QT) % NHEAD;
  const int b  = wave / (QT * NHEAD);

  const int m     = lane & 15;
  const int kp    = (lane >> 4) << 1;
  const int rbase = (lane >> 4) << 3;

  const size_t bn = (size_t)b * NHEAD + n;

  const float* Qrow = QP + (bn * SS + qt * 16 + m) * HH;
  v2f qf[16];
#pragma unroll
  for (int h4 = 0; h4 < 16; ++h4)
    qf[h4] = *(const v2f*)(Qrow + h4 * 4 + kp);

  v8f accv[4] = {{}, {}, {}, {}};
  for (int kt = 0; kt < SS / 16; ++kt) {
    // --- score tile (16q x 16k) ---
    const float* Krow = KP + (bn * SS + kt * 16 + m) * HH;
    v8f s = {};
#pragma unroll
    for (int h4 = 0; h4 < 16; ++h4) {
      v2f bb = *(const v2f*)(Krow + h4 * 4 + kp);
      s = wmma_f32(qf[h4], bb, s);
    }
    // --- normalize per key column (lane%16); park in LDS as P[q][k] ---
    const float cmax = colmax[bn * SS + kt * 16 + m];
    const float cinv = 1.0f / colsum[bn * SS + kt * 16 + m];
#pragma unroll
    for (int r = 0; r < 8; ++r)
      lds[w][(rbase + r) * 18 + m] = __expf(s[r] * 0.125f - cmax) * cinv;
    // same-wave LDS ops are in-order; DScnt dependency handled by compiler.

    // --- P (16x16) x V (16x64): 4 k-chunks x 4 v-tiles ---
    const float* Vb = VPT + (bn * HH) * SS + kt * 16;
#pragma unroll
    for (int kc = 0; kc < 4; ++kc) {
      v2f a = *(const v2f*)(&lds[w][m * 18 + kc * 4 + kp]);
#pragma unroll
      for (int vt = 0; vt < 4; ++vt) {
        v2f bb = *(const v2f*)(Vb + (size_t)(vt * 16 + m) * SS + kc * 4 + kp);
        accv[vt] = wmma_f32(a, bb, accv[vt]);
      }
    }
  }
  float* Orow = HS + ((size_t)b * SS + qt * 16) * EE + n * HH;
#pragma unroll
  for (int vt = 0; vt < 4; ++vt)
#pragma unroll
    for (int r = 0; r < 8; ++r)
      Orow[(size_t)(rbase + r) * EE + vt * 16 + m] = accv[vt][r];
}

// ---------------------------------------------------------------------------
// Kernel 4: output projection  Out[r,e] = sum_j HS[r,j]*Wo[j,e] + bias_o[e]
// [4096 x 1024] x [1024 x 1024], 64x64 block per wave (4x4 tiles).
// ---------------------------------------------------------------------------
__global__ __launch_bounds__(256) void mha_oproj_kernel(
    const float* __restrict__ HS, const float* __restrict__ Wo,
    const float* __restrict__ bo, float* __restrict__ Out)
{
  const int wave = (blockIdx.x * blockDim.x + threadIdx.x) >> 5;
  const int lane = threadIdx.x & 31;
  const int CB = EE / 64;              // 16 col-blocks
  const int cb = wave % CB;
  const int rb = wave / CB;            // 0..63

  const int m     = lane & 15;
  const int kp    = (lane >> 4) << 1;
  const int rbase = (lane >> 4) << 3;

  const float* Arow0 = HS + ((size_t)rb * 64 + m) * EE;
  const float* Bbase = Wo + cb * 64 + m;

  v8f acc[4][4] = {};
  for (int e = 0; e < EE; e += 4) {
    v2f a[4];
#pragma unroll
    for (int rt = 0; rt < 4; ++rt)
      a[rt] = *(const v2f*)(Arow0 + (size_t)rt * 16 * EE + e + kp);
#pragma unroll
    for (int ct = 0; ct < 4; ++ct) {
      v2f bb;
      bb.x = Bbase[(size_t)(e + kp) * EE + ct * 16];
      bb.y = Bbase[(size_t)(e + kp + 1) * EE + ct * 16];
#pragma unroll
      for (int rt = 0; rt < 4; ++rt)
        acc[rt][ct] = wmma_f32(a[rt], bb, acc[rt][ct]);
    }
  }
  float* Obase = Out + ((size_t)rb * 64) * EE + cb * 64 + m;
#pragma unroll
  for (int rt = 0; rt < 4; ++rt)
#pragma unroll
    for (int ct = 0; ct < 4; ++ct) {
      const float bval = bo[cb * 64 + ct * 16 + m];
#pragma unroll
      for (int r = 0; r < 8; ++r)
        Obase[(size_t)(rt * 16 + rbase + r) * EE + ct * 16] =
            acc[rt][ct][r] + bval;
    }
}

// ---------------------------------------------------------------------------
extern "C" void kernel_launch(void* const* d_in, const int* in_sizes, int n_in,
                              void* d_out, int out_size, void* d_ws, size_t ws_size,
                              hipStream_t stream) {
  const float* q  = (const float*)d_in[0];
  const float* k  = (const float*)d_in[1];
  const float* v  = (const float*)d_in[2];
  const float* wq = (const float*)d_in[3];
  const float* wk = (const float*)d_in[4];
  const float* wv = (const float*)d_in[5];
  const float* wo = (const float*)d_in[6];
  const float* bq = (const float*)d_in[7];
  const float* bk = (const float*)d_in[8];
  const float* bv = (const float*)d_in[9];
  const float* bo = (const float*)d_in[10];
  float* out = (float*)d_out;

  float* ws = (float*)d_ws;
  const size_t PSZ = (size_t)BB * NHEAD * SS * HH;   // 4,194,304 floats
  float* QP   = ws;
  float* KP   = ws + PSZ;
  float* VPT  = ws + 2 * PSZ;                        // transposed [B,N,H,S]
  float* HS   = ws + 3 * PSZ;                        // [B,S,E] == PSZ floats
  float* cmax = ws + 4 * PSZ;
  float* csum = cmax + (size_t)BB * NHEAD * SS;

  const dim3 blk(256);
  // proj waves: B*N*(S/64) = 1024 -> 128 blocks of 8 waves
  mha_proj_kernel<<<128, blk, 0, stream>>>(q, wq, bq, QP, 0);
  mha_proj_kernel<<<128, blk, 0, stream>>>(k, wk, bk, KP, 0);
  mha_proj_kernel<<<128, blk, 0, stream>>>(v, wv, bv, VPT, 1);
  // B*N*(S/16) = 4096 waves -> 512 blocks
  mha_softmax_stats_kernel<<<512, blk, 0, stream>>>(QP, KP, cmax, csum);
  mha_heads_kernel<<<512, blk, 0, stream>>>(QP, KP, VPT, cmax, csum, HS);
  // oproj waves: (4096/64)*(1024/64) = 1024 -> 128 blocks
  mha_oproj_kernel<<<128, blk, 0, stream>>>(HS, wo, bo, out);
}